// HungarianMSELoss_5446018531489
// MI455X (gfx1250) — compile-verified
//
#include <hip/hip_runtime.h>

// ---------------------------------------------------------------------------
// Fused Hungarian-matching MSE loss for MI455X (gfx1250, wave32, WMMA).
//
//  Phase 1: per-batch G = X·Yᵀ via V_WMMA_F32_16X16X4_F32 (fp32-exact path;
//           GEMM is ~8.6 GFLOP total -> compute is free, precision matters
//           because the LAP result depends on cost *differences*).
//           LAP is invariant to row/col offsets, so cost' = -G replaces
//           ||x||^2 + ||y||^2 - 2 x·y.  Cost tile (128x128 f32 = 64KB) is
//           written straight into LDS (320KB/WGP makes this possible).
//  Phase 2: Jonker-Volgenant shortest augmenting path solver run by wave 0
//           out of LDS; 4 columns per lane, wave-wide min via __shfl_xor.
//  Phase 3: MSE partial per block -> workspace; deterministic fixed-order
//           tree reduction kernel produces the scalar mean.
// ---------------------------------------------------------------------------

typedef __attribute__((ext_vector_type(2))) float v2f;
typedef __attribute__((ext_vector_type(8))) float v8f;

#define KQ 128         // rows/cols of the assignment problem
#define DQ 512         // feature dim
#define KC 64          // D-chunk staged in LDS per iteration
#define XS 68          // padded LDS row stride (floats): 4r mod 64 spreads banks
#define NT 256         // threads per block (8 wave32)
#define WAVES 8
#define INFF 3.0e38f

// LDS layout (bytes)
#define OFF_COST 0
#define SZ_COST  (KQ * KQ * 4)            // 65536
#define OFF_XC   (OFF_COST + SZ_COST)
#define SZ_XC    (KQ * XS * 4)            // 34816
#define OFF_YC   (OFF_XC + SZ_XC)
#define OFF_U    (OFF_YC + SZ_XC)         // 135168
#define OFF_P    (OFF_U   + 132 * 4)
#define OFF_WAY  (OFF_P   + 132 * 4)
#define OFF_PERM (OFF_WAY + 132 * 4)
#define OFF_RED  (OFF_PERM + KQ * 4)
#define SMEM_TOTAL (OFF_RED + WAVES * 4)  // ~137.3 KB < 160 KB

__global__ __launch_bounds__(NT) void hung_fused(const float* __restrict__ xg,
                                                 const float* __restrict__ yg,
                                                 float* __restrict__ partial) {
  extern __shared__ char smem[];
  float* costL = (float*)(smem + OFF_COST);
  float* XcL   = (float*)(smem + OFF_XC);
  float* YcL   = (float*)(smem + OFF_YC);
  float* uL    = (float*)(smem + OFF_U);
  int*   pL    = (int*)  (smem + OFF_P);
  int*   wayL  = (int*)  (smem + OFF_WAY);
  int*   permL = (int*)  (smem + OFF_PERM);
  float* redL  = (float*)(smem + OFF_RED);

  const int tid  = threadIdx.x;
  const int wave = tid >> 5;
  const int lane = tid & 31;
  const size_t base = (size_t)blockIdx.x * (KQ * DQ);
  const float* X = xg + base;
  const float* Y = yg + base;

  // ------------------- Phase 1: cost' = -X·Yᵀ via f32 WMMA -----------------
  // Wave w owns output tile-row w (16 rows); 8 tiles of 16x16 across columns.
  v8f acc[8];
#pragma unroll
  for (int t = 0; t < 8; ++t) {
    v8f z = {0.f, 0.f, 0.f, 0.f, 0.f, 0.f, 0.f, 0.f};
    acc[t] = z;
  }

  const int arow  = (wave << 4) + (lane & 15);  // A-fragment row (M)
  const int khalf = (lane >> 4) << 1;           // K sub-offset per half-wave

  for (int k0 = 0; k0 < DQ; k0 += KC) {
    __syncthreads();  // previous chunk fully consumed before restaging
    // Coalesced stage of X[:,k0:k0+64] and Y[:,k0:k0+64] into padded LDS.
#pragma unroll
    for (int it = 0; it < 8; ++it) {
      int idx = tid + NT * it;        // 0..2047 float4 slots
      int row = idx >> 4;             // 16 float4 per 64-wide row
      int c4  = (idx & 15) << 2;
      *(float4*)(XcL + row * XS + c4) = *(const float4*)(X + row * DQ + k0 + c4);
      *(float4*)(YcL + row * XS + c4) = *(const float4*)(Y + row * DQ + k0 + c4);
    }
    __syncthreads();

    for (int kk = 0; kk < KC; kk += 4) {
      // A 16x4 f32 fragment: lane m holds K=khalf, khalf+1 (8B-aligned b64).
      v2f a = *(const v2f*)(XcL + arow * XS + kk + khalf);
#pragma unroll
      for (int tn = 0; tn < 8; ++tn) {
        int brow = (tn << 4) + (lane & 15);   // B mirrors A with N on lanes
        v2f b = *(const v2f*)(YcL + brow * XS + kk + khalf);
        acc[tn] = __builtin_amdgcn_wmma_f32_16x16x4_f32(
            /*neg_a=*/false, a, /*neg_b=*/false, b,
            /*c_mod=*/(short)0, acc[tn], /*reuse_a=*/false, /*reuse_b=*/false);
      }
    }
  }

  // C/D layout: VGPR r -> M=r (lanes 0-15) / M=r+8 (lanes 16-31), N=lane&15.
  {
    int crow0 = (wave << 4) + ((lane >> 4) << 3);
    int ccol  = lane & 15;
#pragma unroll
    for (int tn = 0; tn < 8; ++tn) {
      int col = (tn << 4) + ccol;
#pragma unroll
      for (int r = 0; r < 8; ++r)
        costL[(crow0 + r) * KQ + col] = -acc[tn][r];
    }
  }
  __syncthreads();

  // ------------------- Phase 2: JV LAP solver (wave 0 only) ----------------
  if (wave == 0) {
    for (int t = lane; t < 132; t += 32) { uL[t] = 0.f; pL[t] = 0; wayL[t] = 0; }
    float vv[4] = {0.f, 0.f, 0.f, 0.f};   // v[4*lane+1 .. 4*lane+4]

    for (int i = 1; i <= KQ; ++i) {
      pL[0] = i;                           // uniform store
      int j0 = 0;
      bool used0 = false;
      unsigned um = 0u;                    // per-lane used bits for 4 columns
      float mv[4] = {INFF, INFF, INFF, INFF};

      for (int step = 0; step < 2 * KQ + 8; ++step) {
        if (j0 == 0) used0 = true;
        else {
          int jj = j0 - 1;
          if ((jj >> 2) == lane) um |= (1u << (jj & 3));
        }
        int   i0  = pL[j0];                // LDS broadcast (uniform address)
        float ui0 = uL[i0];
        const float* crow = costL + (i0 - 1) * KQ;
#pragma unroll
        for (int t = 0; t < 4; ++t) {
          if (!((um >> t) & 1u)) {
            int jj = (lane << 2) + t;
            float cur = crow[jj] - ui0 - vv[t];
            if (cur < mv[t]) { mv[t] = cur; wayL[jj + 1] = j0; }
          }
        }
        // wave-wide argmin over free columns (first-index tie-break)
        float best = INFF; int bj = 0;
#pragma unroll
        for (int t = 0; t < 4; ++t)
          if (!((um >> t) & 1u) && mv[t] < best) { best = mv[t]; bj = (lane << 2) + t + 1; }
        for (int off = 16; off > 0; off >>= 1) {
          float ob = __shfl_xor(best, off, 32);
          int   oj = __shfl_xor(bj,   off, 32);
          if (oj != 0 && (ob < best || (ob == best && (bj == 0 || oj < bj)))) {
            best = ob; bj = oj;
          }
        }
        float delta = best;
#pragma unroll
        for (int t = 0; t < 4; ++t) {
          if ((um >> t) & 1u) {
            int pj = pL[(lane << 2) + t + 1];
            uL[pj] += delta;               // matched rows are distinct: no races
            vv[t]  -= delta;
          } else {
            mv[t]  -= delta;
          }
        }
        if (used0 && lane == 0) uL[pL[0]] += delta;   // u[i] += delta
        j0 = bj;
        if (pL[j0] == 0) break;
      }
      // augment along alternating path (wave-uniform scalar chain)
      while (j0 != 0) {
        int jn = wayL[j0];
        pL[j0] = pL[jn];
        j0 = jn;
      }
    }
    for (int j = lane + 1; j <= KQ; j += 32)
      permL[pL[j] - 1] = j - 1;            // col_ind[p[j]-1] = j-1
  }
  __syncthreads();

  // ------------------- Phase 3: block-local MSE partial --------------------
  float s = 0.f;
#pragma unroll 4
  for (int it = 0; it < (KQ * DQ) / NT; ++it) {
    int idx = tid + NT * it;
    int row = idx >> 9;                    // /512 (uniform per iteration)
    int col = idx & (DQ - 1);
    float d = X[row * DQ + col] - Y[permL[row] * DQ + col];
    s += d * d;
  }
  for (int off = 16; off > 0; off >>= 1) s += __shfl_xor(s, off, 32);
  if (lane == 0) redL[wave] = s;
  __syncthreads();
  if (tid == 0) {
    float tot = 0.f;
    for (int w = 0; w < WAVES; ++w) tot += redL[w];
    partial[blockIdx.x] = tot;
  }
}

// Deterministic fixed-order final reduction (no float atomics).
__global__ __launch_bounds__(256) void hung_finalize(const float* __restrict__ partial,
                                                     float* __restrict__ out,
                                                     int nb, float scale) {
  __shared__ float sm[256];
  float s = 0.f;
  for (int i = threadIdx.x; i < nb; i += 256) s += partial[i];
  sm[threadIdx.x] = s;
  __syncthreads();
  for (int st = 128; st > 0; st >>= 1) {
    if (threadIdx.x < st) sm[threadIdx.x] += sm[threadIdx.x + st];
    __syncthreads();
  }
  if (threadIdx.x == 0) out[0] = sm[0] * scale;
}

extern "C" void kernel_launch(void* const* d_in, const int* in_sizes, int n_in,
                              void* d_out, int out_size, void* d_ws, size_t ws_size,
                              hipStream_t stream) {
  const float* x = (const float*)d_in[0];
  const float* y = (const float*)d_in[1];
  float* out     = (float*)d_out;
  float* partial = (float*)d_ws;           // nb floats of scratch

  const int nb = in_sizes[0] / (KQ * DQ);  // 512 batches for the reference
  hung_fused<<<nb, NT, SMEM_TOTAL, stream>>>(x, y, partial);
  const float scale = 1.0f / ((float)nb * (float)KQ * (float)DQ);
  hung_finalize<<<1, 256, 0, stream>>>(partial, out, nb, scale);
}